// DecagonModel_17910013624847
// MI455X (gfx1250) — compile-verified
//
#include <hip/hip_runtime.h>

typedef float v2f __attribute__((ext_vector_type(2)));
typedef float v8f __attribute__((ext_vector_type(8)));

// ---------------- problem constants (match reference) ----------------
constexpr int N0 = 20000, N1 = 2000;
constexpr int FF = 512, HH = 256, DD = 128, KK = 16;
constexpr int E00 = 640000, E01 = 100000, E10 = 100000, E11 = 64000;

// ---------------- workspace layout (float offsets) ----------------
constexpr long long off_xw00 = 0;                       // 5,120,000  (-> h0)
constexpr long long off_xw10 = 5120000;                 // 5,120,000  (-> hw00/hw10)
constexpr long long off_xw01 = 10240000;                //   512,000  (-> h1)
constexpr long long off_xw11 = 10752000;                // 8,192,000  (-> hw01/hw11)
constexpr long long off_acc00 = 18944000;               // 5,120,000  (-> bcc00/bcc01)
constexpr long long off_acc01 = 24064000;               // 5,120,000  (-> z0R00/z0R01)
constexpr long long off_acc10 = 29184000;               //   512,000  (-> bcc10 ; z1R10)
constexpr long long off_acc11 = 29696000;               // 8,192,000  (-> bcc11 / z1d / Zk)
// total = 37,888,000 floats = 151,552,000 bytes

constexpr long long off_h0    = off_xw00;
constexpr long long off_h1    = off_xw01;
constexpr long long off_hw00  = off_xw10;
constexpr long long off_hw10  = off_xw10 + 2560000;
constexpr long long off_hw01  = off_xw11;
constexpr long long off_hw11  = off_xw11 + 256000;
constexpr long long off_bcc00 = off_acc00;
constexpr long long off_bcc01 = off_acc00 + 2560000;
constexpr long long off_bcc10 = off_acc10;
constexpr long long off_bcc11 = off_acc11;
constexpr long long off_z0R00 = off_acc01;
constexpr long long off_z0R01 = off_acc01 + 2560000;
constexpr long long off_z1R10 = off_acc10 + 256000;
constexpr long long off_z1d   = off_acc11 + 4096000;
constexpr long long off_Zk    = off_acc11;

__device__ __forceinline__ float wave_sum(float v) {
#pragma unroll
  for (int off = 16; off > 0; off >>= 1) v += __shfl_xor(v, off, 32);
  return v;
}

// ---------------- batched GEMM: C = A @ B (fp32 WMMA 16x16x4, LDS double-buffered) ----
// Block tile 128x64 (8 waves, wave = 16x64); K chunk = 32. M%16==0, N%64==0, K%32==0.
// A tile: row-major, padded stride 36 (16-row fragment reads are bank-conflict-free).
// B tile: k-pair interleaved: Bs[p*BPS + col*2 + (k&1)], p = k>>1. One WMMA B-fragment
//         is then a single aligned 8-byte LDS load. BPS=160 (==32 mod 64) puts the two
//         lane-halves (pairs p, p+1) on complementary 32-bank sets -> conflict-free.
constexpr int BK   = 32;
constexpr int ASTR = 36;
constexpr int BPS  = 160;

__global__ __launch_bounds__(256) void gemm_f32_wmma(
    const float* __restrict__ A, const float* __restrict__ B, float* __restrict__ C,
    int M, int N, int Kd, long long sA, long long sB, long long sC) {
  __shared__ float As[2][128 * ASTR];      // 36,864 B
  __shared__ float Bs[2][(BK / 2) * BPS];  // 20,480 B

  const int b = blockIdx.z;
  A += (long long)b * sA;
  B += (long long)b * sB;
  C += (long long)b * sC;

  const int tid  = threadIdx.x;
  const int lane = tid & 31;
  const int wave = tid >> 5;
  const int mBlk = blockIdx.x * 128;
  const int n0   = blockIdx.y * 64;
  const int m0   = mBlk + wave * 16;
  const int half = lane >> 4;   // 0: lanes 0-15 (K 0/1) ; 1: lanes 16-31 (K 2/3)
  const int lr   = lane & 15;
  const bool active = (m0 < M);  // wave-uniform

  // cooperative fill indices
  const int arow = tid >> 3;          // 0..31  (x4 passes -> 128 rows)
  const int ac4  = (tid & 7) * 4;     // 0..28
  const int brow = tid >> 4;          // 0..15  (x2 passes -> 32 rows)
  const int bc4  = (tid & 15) * 4;    // 0..60

  const int nChunks = Kd / BK;
  const float4 zero4 = make_float4(0.f, 0.f, 0.f, 0.f);

  // ---- prologue: fill chunk 0 ----
  {
#pragma unroll
    for (int p = 0; p < 4; ++p) {
      const int row = arow + p * 32;
      const int grow = mBlk + row;
      float4 v = (grow < M) ? *(const float4*)(A + (long long)grow * Kd + ac4) : zero4;
      *(float4*)&As[0][row * ASTR + ac4] = v;
    }
#pragma unroll
    for (int p = 0; p < 2; ++p) {
      const int row = brow + p * 16;
      const float4 v = *(const float4*)(B + (long long)row * N + n0 + bc4);
      float* bp = &Bs[0][(row >> 1) * BPS + bc4 * 2 + (row & 1)];
      bp[0] = v.x; bp[2] = v.y; bp[4] = v.z; bp[6] = v.w;
    }
  }
  __syncthreads();

  v8f acc0 = {}, acc1 = {}, acc2 = {}, acc3 = {};

  for (int c = 0; c < nChunks; ++c) {
    const int buf = c & 1;
    const bool pf = (c + 1 < nChunks);
    float4 ra[4], rb[2];

    // ---- prefetch next chunk into registers (overlaps with WMMA below) ----
    if (pf) {
      const int k0 = (c + 1) * BK;
#pragma unroll
      for (int p = 0; p < 4; ++p) {
        const int grow = mBlk + arow + p * 32;
        ra[p] = (grow < M) ? *(const float4*)(A + (long long)grow * Kd + k0 + ac4) : zero4;
      }
#pragma unroll
      for (int p = 0; p < 2; ++p)
        rb[p] = *(const float4*)(B + (long long)(k0 + brow + p * 16) * N + n0 + bc4);
    }

    // ---- compute on current chunk from LDS ----
    if (active) {
      const float* Asb = &As[buf][(wave * 16 + lr) * ASTR + 2 * half];
      const float* Bsb = &Bs[buf][half * BPS + lr * 2];
#pragma unroll
      for (int kk = 0; kk < BK; kk += 4) {
        const v2f a = *(const v2f*)(Asb + kk);
        const float* bp = Bsb + (kk >> 1) * BPS;
        const v2f b0 = *(const v2f*)(bp);
        const v2f b1 = *(const v2f*)(bp + 32);
        const v2f b2 = *(const v2f*)(bp + 64);
        const v2f b3 = *(const v2f*)(bp + 96);
        acc0 = __builtin_amdgcn_wmma_f32_16x16x4_f32(false, a, false, b0, (short)0, acc0, false, false);
        acc1 = __builtin_amdgcn_wmma_f32_16x16x4_f32(false, a, false, b1, (short)0, acc1, false, false);
        acc2 = __builtin_amdgcn_wmma_f32_16x16x4_f32(false, a, false, b2, (short)0, acc2, false, false);
        acc3 = __builtin_amdgcn_wmma_f32_16x16x4_f32(false, a, false, b3, (short)0, acc3, false, false);
      }
    }
    __syncthreads();

    // ---- store prefetched regs into the other LDS buffer ----
    if (pf) {
      const int nb = buf ^ 1;
#pragma unroll
      for (int p = 0; p < 4; ++p)
        *(float4*)&As[nb][(arow + p * 32) * ASTR + ac4] = ra[p];
#pragma unroll
      for (int p = 0; p < 2; ++p) {
        const int row = brow + p * 16;
        float* bp = &Bs[nb][(row >> 1) * BPS + bc4 * 2 + (row & 1)];
        bp[0] = rb[p].x; bp[2] = rb[p].y; bp[4] = rb[p].z; bp[6] = rb[p].w;
      }
      __syncthreads();
    }
  }

  // ---- store C : VGPR r -> row r (lanes 0-15) / r+8 (lanes 16-31), col = lane&15 ----
  if (active) {
#pragma unroll
    for (int r = 0; r < 8; ++r) {
      const int row = m0 + r + 8 * half;
      float* cp = C + (long long)row * N + n0 + lr;
      cp[0]  = acc0[r];
      cp[16] = acc1[r];
      cp[32] = acc2[r];
      cp[48] = acc3[r];
    }
  }
}

// ---------------- SPMM: out[dst] += val * x[src]  (wave per edge, float4 gathers) ----
__global__ void spmm_edges(const int* __restrict__ dst, const int* __restrict__ src,
                           const float* __restrict__ val, const float* __restrict__ x,
                           float* __restrict__ out, int nE, int N,
                           int eStride, long long xStride, long long oStride) {
  const int k = blockIdx.y;
  dst += (long long)k * eStride;
  src += (long long)k * eStride;
  val += (long long)k * eStride;
  x   += (long long)k * xStride;
  out += (long long)k * oStride;
  const int w = blockIdx.x * (blockDim.x >> 5) + (threadIdx.x >> 5);
  if (w >= nE) return;
  const int lane = threadIdx.x & 31;
  const int d = dst[w];
  const int s = src[w];
  const float v = val[w];
  const float4* xs = (const float4*)(x + (long long)s * N);
  float* od = out + (long long)d * N;
  const int nq = N >> 2;
  for (int q = lane; q < nq; q += 32) {
    const float4 t = xs[q];
    float* o = od + q * 4;
    atomicAdd(o + 0, v * t.x);
    atomicAdd(o + 1, v * t.y);
    atomicAdd(o + 2, v * t.z);
    atomicAdd(o + 3, v * t.w);
  }
}

// ---------------- conv finalize: out = l2norm( sum_k act(buf_k) ), wave per row ----
__global__ void finalize_conv(const float* __restrict__ buf, float* __restrict__ out,
                              int K, int n, int N, long long kStride, int do_relu) {
  const int wave = threadIdx.x >> 5;
  const int lane = threadIdx.x & 31;
  const int row = blockIdx.x * (blockDim.x >> 5) + wave;
  if (row >= n) return;
  const int nc = N >> 5;  // 8 (N=256) or 4 (N=128)
  float vals[8];
  float ss = 0.f;
  for (int j = 0; j < nc; ++j) {
    const int c = lane + (j << 5);
    float a = 0.f;
    for (int k = 0; k < K; ++k) {
      float t = buf[(long long)k * kStride + (long long)row * N + c];
      if (do_relu) t = fmaxf(t, 0.f);
      a += t;
    }
    vals[j] = a;
    ss += a * a;
  }
  ss = wave_sum(ss);
  const float sc = rsqrtf(fmaxf(ss, 1e-12f));
  for (int j = 0; j < nc; ++j)
    out[(long long)row * N + lane + (j << 5)] = vals[j] * sc;
}

// ---------------- elementwise: o = act(a + b) ----------------
__global__ void add_act(const float* __restrict__ a, const float* __restrict__ b,
                        float* __restrict__ o, long long n, int do_relu) {
  const long long i = (long long)blockIdx.x * blockDim.x + threadIdx.x;
  if (i >= n) return;
  float v = a[i] + b[i];
  o[i] = do_relu ? fmaxf(v, 0.f) : v;
}

// ---------------- z1d[k][i][f] = z1[i][f] * dloc[k][f] ----------------
__global__ void scale_dloc(const float* __restrict__ z1, const float* __restrict__ dloc,
                           float* __restrict__ out, long long total) {
  const long long i = (long long)blockIdx.x * blockDim.x + threadIdx.x;
  if (i >= total) return;
  const int f = (int)(i % DD);
  const long long rk = i / DD;
  const int r = (int)(rk % N1);
  const int k = (int)(rk / N1);
  out[i] = z1[(long long)r * DD + f] * dloc[k * DD + f];
}

// ---------------- edge scores: out[e] = dot(zl[dst[e]], zr[src[e]]), D=128 ----------------
__global__ void edge_dot128(const float* __restrict__ zl, const float* __restrict__ zr,
                            const int* __restrict__ dst, const int* __restrict__ src,
                            float* __restrict__ out, int nE, int eStride,
                            long long zlStride, long long zrStride, long long oStride) {
  const int k = blockIdx.y;
  zl  += (long long)k * zlStride;
  zr  += (long long)k * zrStride;
  dst += (long long)k * eStride;
  src += (long long)k * eStride;
  out += (long long)k * oStride;
  const int e = blockIdx.x * (blockDim.x >> 5) + (threadIdx.x >> 5);
  if (e >= nE) return;
  const int lane = threadIdx.x & 31;
  const int d = dst[e];
  const int s = src[e];
  const float4 a = ((const float4*)(zl + (long long)d * DD))[lane];
  const float4 b = ((const float4*)(zr + (long long)s * DD))[lane];
  float acc = a.x * b.x + a.y * b.y + a.z * b.z + a.w * b.w;
  acc = wave_sum(acc);
  if (lane == 0) out[e] = acc;
}

// ---------------- launch ----------------
extern "C" void kernel_launch(void* const* d_in, const int* in_sizes, int n_in,
                              void* d_out, int out_size, void* d_ws, size_t ws_size,
                              hipStream_t stream) {
  (void)in_sizes; (void)n_in; (void)out_size; (void)ws_size;

  const float* feat0   = (const float*)d_in[0];
  const float* feat1   = (const float*)d_in[1];
  const int*   e00_dst = (const int*)d_in[2];
  const int*   e00_src = (const int*)d_in[3];
  const float* e00_val = (const float*)d_in[4];
  const int*   e01_dst = (const int*)d_in[5];
  const int*   e01_src = (const int*)d_in[6];
  const float* e01_val = (const float*)d_in[7];
  const int*   e10_dst = (const int*)d_in[8];
  const int*   e10_src = (const int*)d_in[9];
  const float* e10_val = (const float*)d_in[10];
  const int*   e11_dst = (const int*)d_in[11];
  const int*   e11_src = (const int*)d_in[12];
  const float* e11_val = (const float*)d_in[13];
  const float* W1_00 = (const float*)d_in[14];
  const float* W1_01 = (const float*)d_in[15];
  const float* W1_10 = (const float*)d_in[16];
  const float* W1_11 = (const float*)d_in[17];
  const float* W2_00 = (const float*)d_in[18];
  const float* W2_01 = (const float*)d_in[19];
  const float* W2_10 = (const float*)d_in[20];
  const float* W2_11 = (const float*)d_in[21];
  const float* R00   = (const float*)d_in[22];
  const float* R01   = (const float*)d_in[23];
  const float* R10   = (const float*)d_in[24];
  const float* Rglob = (const float*)d_in[25];
  const float* dloc  = (const float*)d_in[26];

  float* ws = (float*)d_ws;
  float* out = (float*)d_out;
  float* z0  = out;                       // 20000*128
  float* z1  = out + 2560000;             // 2000*128
  float* s00 = z1 + 256000;               // 640000
  float* s01 = s00 + E00;                 // 100000
  float* s10 = s01 + E01;                 // 100000
  float* s11 = s10 + E10;                 // 16*64000

  const dim3 blk(256);
  auto gg = [](int M, int N, int batch) { return dim3((M + 127) / 128, N / 64, batch); };

  // ---- layer 1 dense: xW ----
  gemm_f32_wmma<<<gg(N0, HH, 1), blk, 0, stream>>>(feat0, W1_00, ws + off_xw00, N0, HH, FF, 0, 0, 0);
  gemm_f32_wmma<<<gg(N0, HH, 1), blk, 0, stream>>>(feat0, W1_10, ws + off_xw10, N0, HH, FF, 0, 0, 0);
  gemm_f32_wmma<<<gg(N1, HH, 1), blk, 0, stream>>>(feat1, W1_01, ws + off_xw01, N1, HH, FF, 0, 0, 0);
  gemm_f32_wmma<<<gg(N1, HH, KK), blk, 0, stream>>>(feat1, W1_11, ws + off_xw11, N1, HH, FF,
                                                    0, (long long)FF * HH, (long long)N1 * HH);

  // ---- zero layer-1 accumulators (regions E..H contiguous) ----
  hipMemsetAsync(ws + off_acc00, 0, (size_t)18944000 * sizeof(float), stream);

  // ---- layer 1 SPMM ----
  spmm_edges<<<dim3((E00 + 7) / 8, 1), blk, 0, stream>>>(e00_dst, e00_src, e00_val, ws + off_xw00,
                                                         ws + off_acc00, E00, HH, 0, 0, 0);
  spmm_edges<<<dim3((E01 + 7) / 8, 1), blk, 0, stream>>>(e01_dst, e01_src, e01_val, ws + off_xw01,
                                                         ws + off_acc01, E01, HH, 0, 0, 0);
  spmm_edges<<<dim3((E10 + 7) / 8, 1), blk, 0, stream>>>(e10_dst, e10_src, e10_val, ws + off_xw10,
                                                         ws + off_acc10, E10, HH, 0, 0, 0);
  spmm_edges<<<dim3((E11 + 7) / 8, KK), blk, 0, stream>>>(e11_dst, e11_src, e11_val, ws + off_xw11,
                                                          ws + off_acc11, E11, HH, E11,
                                                          (long long)N1 * HH, (long long)N1 * HH);

  // ---- layer 1 conv finalize: l2norm(sum_k relu(.)) (in place) ----
  finalize_conv<<<dim3((N0 + 7) / 8), blk, 0, stream>>>(ws + off_acc00, ws + off_acc00, 1, N0, HH, 0, 1);
  finalize_conv<<<dim3((N0 + 7) / 8), blk, 0, stream>>>(ws + off_acc01, ws + off_acc01, 1, N0, HH, 0, 1);
  finalize_conv<<<dim3((N1 + 7) / 8), blk, 0, stream>>>(ws + off_acc10, ws + off_acc10, 1, N1, HH, 0, 1);
  finalize_conv<<<dim3((N1 + 7) / 8), blk, 0, stream>>>(ws + off_acc11, ws + off_acc11, KK, N1, HH,
                                                        (long long)N1 * HH, 1);

  // ---- hidden: h = relu(conv_a + conv_b) ----
  add_act<<<dim3((N0 * HH + 255) / 256), blk, 0, stream>>>(ws + off_acc00, ws + off_acc01,
                                                           ws + off_h0, (long long)N0 * HH, 1);
  add_act<<<dim3((N1 * HH + 255) / 256), blk, 0, stream>>>(ws + off_acc10, ws + off_acc11,
                                                           ws + off_h1, (long long)N1 * HH, 1);

  // ---- layer 2 dense: hW ----
  gemm_f32_wmma<<<gg(N0, DD, 1), blk, 0, stream>>>(ws + off_h0, W2_00, ws + off_hw00, N0, DD, HH, 0, 0, 0);
  gemm_f32_wmma<<<gg(N0, DD, 1), blk, 0, stream>>>(ws + off_h0, W2_10, ws + off_hw10, N0, DD, HH, 0, 0, 0);
  gemm_f32_wmma<<<gg(N1, DD, 1), blk, 0, stream>>>(ws + off_h1, W2_01, ws + off_hw01, N1, DD, HH, 0, 0, 0);
  gemm_f32_wmma<<<gg(N1, DD, KK), blk, 0, stream>>>(ws + off_h1, W2_11, ws + off_hw11, N1, DD, HH,
                                                    0, (long long)HH * DD, (long long)N1 * DD);

  // ---- zero layer-2 accumulators ----
  hipMemsetAsync(ws + off_bcc00, 0, (size_t)5120000 * sizeof(float), stream);  // bcc00+bcc01
  hipMemsetAsync(ws + off_bcc10, 0, (size_t)256000 * sizeof(float), stream);
  hipMemsetAsync(ws + off_bcc11, 0, (size_t)4096000 * sizeof(float), stream);

  // ---- layer 2 SPMM ----
  spmm_edges<<<dim3((E00 + 7) / 8, 1), blk, 0, stream>>>(e00_dst, e00_src, e00_val, ws + off_hw00,
                                                         ws + off_bcc00, E00, DD, 0, 0, 0);
  spmm_edges<<<dim3((E01 + 7) / 8, 1), blk, 0, stream>>>(e01_dst, e01_src, e01_val, ws + off_hw01,
                                                         ws + off_bcc01, E01, DD, 0, 0, 0);
  spmm_edges<<<dim3((E10 + 7) / 8, 1), blk, 0, stream>>>(e10_dst, e10_src, e10_val, ws + off_hw10,
                                                         ws + off_bcc10, E10, DD, 0, 0, 0);
  spmm_edges<<<dim3((E11 + 7) / 8, KK), blk, 0, stream>>>(e11_dst, e11_src, e11_val, ws + off_hw11,
                                                          ws + off_bcc11, E11, DD, E11,
                                                          (long long)N1 * DD, (long long)N1 * DD);

  // ---- layer 2 conv finalize: l2norm(sum_k identity(.)) (in place) ----
  finalize_conv<<<dim3((N0 + 7) / 8), blk, 0, stream>>>(ws + off_bcc00, ws + off_bcc00, 1, N0, DD, 0, 0);
  finalize_conv<<<dim3((N0 + 7) / 8), blk, 0, stream>>>(ws + off_bcc01, ws + off_bcc01, 1, N0, DD, 0, 0);
  finalize_conv<<<dim3((N1 + 7) / 8), blk, 0, stream>>>(ws + off_bcc10, ws + off_bcc10, 1, N1, DD, 0, 0);
  finalize_conv<<<dim3((N1 + 7) / 8), blk, 0, stream>>>(ws + off_bcc11, ws + off_bcc11, KK, N1, DD,
                                                        (long long)N1 * DD, 0);

  // ---- embeddings z0, z1 (written straight into d_out) ----
  add_act<<<dim3((N0 * DD + 255) / 256), blk, 0, stream>>>(ws + off_bcc00, ws + off_bcc01, z0,
                                                           (long long)N0 * DD, 0);
  add_act<<<dim3((N1 * DD + 255) / 256), blk, 0, stream>>>(ws + off_bcc10, ws + off_bcc11, z1,
                                                           (long long)N1 * DD, 0);

  // ---- decoder precomputes: zR = z @ R (turns per-edge D^2 into D) ----
  gemm_f32_wmma<<<gg(N0, DD, 1), blk, 0, stream>>>(z0, R00, ws + off_z0R00, N0, DD, DD, 0, 0, 0);
  gemm_f32_wmma<<<gg(N0, DD, 1), blk, 0, stream>>>(z0, R01, ws + off_z0R01, N0, DD, DD, 0, 0, 0);
  gemm_f32_wmma<<<gg(N1, DD, 1), blk, 0, stream>>>(z1, R10, ws + off_z1R10, N1, DD, DD, 0, 0, 0);

  // DEDICOM: z1d_k = z1 * d_k ; Z_k = z1d_k @ Rglob
  scale_dloc<<<dim3((KK * N1 * DD + 255) / 256), blk, 0, stream>>>(z1, dloc, ws + off_z1d,
                                                                   (long long)KK * N1 * DD);
  gemm_f32_wmma<<<gg(N1, DD, KK), blk, 0, stream>>>(ws + off_z1d, Rglob, ws + off_Zk, N1, DD, DD,
                                                    (long long)N1 * DD, 0, (long long)N1 * DD);

  // ---- per-edge scores ----
  edge_dot128<<<dim3((E00 + 7) / 8, 1), blk, 0, stream>>>(ws + off_z0R00, z0, e00_dst, e00_src,
                                                          s00, E00, 0, 0, 0, 0);
  edge_dot128<<<dim3((E01 + 7) / 8, 1), blk, 0, stream>>>(ws + off_z0R01, z1, e01_dst, e01_src,
                                                          s01, E01, 0, 0, 0, 0);
  edge_dot128<<<dim3((E10 + 7) / 8, 1), blk, 0, stream>>>(ws + off_z1R10, z0, e10_dst, e10_src,
                                                          s10, E10, 0, 0, 0, 0);
  edge_dot128<<<dim3((E11 + 7) / 8, KK), blk, 0, stream>>>(ws + off_Zk, ws + off_z1d, e11_dst, e11_src,
                                                           s11, E11, E11,
                                                           (long long)N1 * DD, (long long)N1 * DD,
                                                           (long long)E11);
}